// SparseConv2D_26070451487302
// MI455X (gfx1250) — compile-verified
//
#include <hip/hip_runtime.h>

typedef float v2f __attribute__((ext_vector_type(2)));
typedef float v4f __attribute__((ext_vector_type(4)));
typedef float v8f __attribute__((ext_vector_type(8)));
typedef int   v4i __attribute__((ext_vector_type(4)));

#if __has_builtin(__builtin_amdgcn_global_load_async_to_lds_b128) && \
    __has_builtin(__builtin_amdgcn_global_load_async_to_lds_b32)
#define ASYNC_LDS 1
#else
#define ASYNC_LDS 0
#endif

namespace {
constexpr int C_IN  = 128;
constexpr int H_IN  = 56;
constexpr int W_IN  = 56;
constexpr int HW    = H_IN * W_IN;           // 3136
constexpr int F_OUT = 256;
constexpr int KSZ   = 3;
constexpr int KDIM  = C_IN * KSZ * KSZ;      // 1152
constexpr int OSP   = 54;
constexpr int P_IMG = OSP * OSP;             // 2916
constexpr int OUT_IMG = F_OUT * P_IMG;

constexpr int M_BLK = 128;                   // filters per block
constexpr int P_BLK = 64;                    // pixels per block
constexpr int K_BLK = 16;                    // K per stage
constexpr int LSTR  = 20;                    // padded LDS row stride (floats)
constexpr int KTILES = KDIM / K_BLK;         // 72
constexpr int NBUF  = 3;                     // triple-buffer for async DMA
}

__device__ __forceinline__ void async_g2l_b128(const float* g, float* l) {
#if ASYNC_LDS
    __builtin_amdgcn_global_load_async_to_lds_b128(
        (__attribute__((address_space(1))) v4i*)(g),
        (__attribute__((address_space(3))) v4i*)(l), 0, 0);
#else
    (void)g; (void)l;
#endif
}
__device__ __forceinline__ void async_g2l_b32(const float* g, float* l) {
#if ASYNC_LDS
    __builtin_amdgcn_global_load_async_to_lds_b32(
        (__attribute__((address_space(1))) int*)(g),
        (__attribute__((address_space(3))) int*)(l), 0, 0);
#else
    (void)g; (void)l;
#endif
}
__device__ __forceinline__ void wait_async_le6() {
#if ASYNC_LDS
#if __has_builtin(__builtin_amdgcn_s_wait_asynccnt)
    __builtin_amdgcn_s_wait_asynccnt(6);
#else
    asm volatile("s_wait_asynccnt 0x6" ::: "memory");
#endif
#endif
}
__device__ __forceinline__ void wait_async_le0() {
#if ASYNC_LDS
#if __has_builtin(__builtin_amdgcn_s_wait_asynccnt)
    __builtin_amdgcn_s_wait_asynccnt(0);
#else
    asm volatile("s_wait_asynccnt 0x0" ::: "memory");
#endif
#endif
}

__global__ __launch_bounds__(256, 2)
void sparse_conv2d_wmma_f32(const float* __restrict__ x,
                            const float* __restrict__ w,
                            float* __restrict__ out)
{
    __shared__ float ldsA[NBUF][M_BLK * LSTR];   // [f][k] k-contiguous, padded
    __shared__ float ldsB[NBUF][P_BLK * LSTR];   // [p][k] k-contiguous, padded

    const int tid  = threadIdx.x;
    const int lane = tid & 31;
    const int wave = tid >> 5;

    const int f0 = blockIdx.x * M_BLK;           // 0..1  * 128
    const int p0 = blockIdx.y * P_BLK;           // 0..45 * 64
    const int n  = blockIdx.z;

    const float* xn = x + (size_t)n * (C_IN * HW);

    // ---- A loader: 256 threads -> 128 rows x 2 oct-quads of K (8 floats ea) ----
    const int arow = tid >> 1;                   // 0..127
    const int akq  = (tid & 1) * 8;              // 0 or 8
    const float* agp = w + (size_t)(f0 + arow) * KDIM + akq;

    // ---- B loader: 256 threads -> 64 pixel rows x 4 k-quads (4 floats ea) ----
    const int brow = tid >> 2;                   // 0..63
    const int bkq  = (tid & 3) * 4;              // 0,4,8,12
    int pg = p0 + brow;
    if (pg >= P_IMG) pg = P_IMG - 1;             // clamp: garbage cols never stored
    const int wo    = pg / OSP;
    const int ho    = pg - wo * OSP;
    const int sbase = ho * W_IN + wo;

#if !ASYNC_LDS
    v4f   avr0, avr1;
    float bv0, bv1, bv2, bv3;
#endif

    auto issue = [&](int kt, int buf) {
        const int kb = kt * K_BLK;
        float* al = &ldsA[buf][arow * LSTR + akq];
        // B gather geometry: 4 consecutive k share (i,j), consecutive c
        const int k0 = kb + bkq;
        const int i  = k0 / (C_IN * KSZ);        // /384
        const int r  = k0 - i * (C_IN * KSZ);
        const int j  = r >> 7;
        const int c0 = r & 127;
        const float* bp = xn + c0 * HW + sbase + j * W_IN + i;
        float* bl = &ldsB[buf][brow * LSTR + bkq];
#if ASYNC_LDS
        async_g2l_b128(agp + kb,     al);
        async_g2l_b128(agp + kb + 4, al + 4);
        async_g2l_b32(bp + 0 * HW, bl + 0);
        async_g2l_b32(bp + 1 * HW, bl + 1);
        async_g2l_b32(bp + 2 * HW, bl + 2);
        async_g2l_b32(bp + 3 * HW, bl + 3);
#else
        avr0 = *(const v4f*)(agp + kb);
        avr1 = *(const v4f*)(agp + kb + 4);
        bv0 = bp[0 * HW]; bv1 = bp[1 * HW]; bv2 = bp[2 * HW]; bv3 = bp[3 * HW];
        *(v4f*)al       = avr0;
        *(v4f*)(al + 4) = avr1;
        v4f bq; bq.x = bv0; bq.y = bv1; bq.z = bv2; bq.w = bv3;
        *(v4f*)bl = bq;
#endif
    };

    // ---- per-wave WMMA tile: 32 (F) x 32 (pixels), 4 accumulators ----
    const int waveM = wave >> 1;                 // 0..3 -> rows waveM*32
    const int waveP = wave & 1;                  // 0..1 -> cols waveP*32
    const int mlane = lane & 15;
    const int kh    = (lane >> 4) * 2;           // lane-half selects K-pair

    v8f acc00 = {0,0,0,0,0,0,0,0}, acc01 = {0,0,0,0,0,0,0,0};
    v8f acc10 = {0,0,0,0,0,0,0,0}, acc11 = {0,0,0,0,0,0,0,0};

    auto compute = [&](int buf) {
        const float* aR = &ldsA[buf][(waveM * 32 + mlane) * LSTR + kh];
        const float* bR = &ldsB[buf][(waveP * 32 + mlane) * LSTR + kh];
#pragma unroll
        for (int ks = 0; ks < 4; ++ks) {
            v2f a0 = *(const v2f*)(aR + ks * 4);
            v2f a1 = *(const v2f*)(aR + 16 * LSTR + ks * 4);
            v2f b0 = *(const v2f*)(bR + ks * 4);
            v2f b1 = *(const v2f*)(bR + 16 * LSTR + ks * 4);
            acc00 = __builtin_amdgcn_wmma_f32_16x16x4_f32(false, a0, false, b0, (short)0, acc00, false, false);
            acc01 = __builtin_amdgcn_wmma_f32_16x16x4_f32(false, a0, false, b1, (short)0, acc01, false, false);
            acc10 = __builtin_amdgcn_wmma_f32_16x16x4_f32(false, a1, false, b0, (short)0, acc10, false, false);
            acc11 = __builtin_amdgcn_wmma_f32_16x16x4_f32(false, a1, false, b1, (short)0, acc11, false, false);
        }
    };

    // ---- pipeline: 3 LDS buffers, 1 barrier/stage, issue-after-barrier ----
    issue(0, 0);
    issue(1, 1);
    int bc = 0;          // compute buffer
    int bi = 2;          // issue buffer
    for (int kt = 0; kt < KTILES; ++kt) {
        if (kt + 1 < KTILES) wait_async_le6();   // stage kt landed (in-order)
        else                 wait_async_le0();
        __syncthreads();                         // all threads' stage-kt data visible
        if (kt + 2 < KTILES) issue(kt + 2, bi);  // safe: last read of bi was kt-1
        compute(bc);
        bc = (bc == NBUF - 1) ? 0 : bc + 1;
        bi = (bi == NBUF - 1) ? 0 : bi + 1;
    }

    // ---- store: D layout: VGPR r, lanes 0-15: M=r, N=lane; lanes 16-31: M=r+8 ----
    float* outn = out + (size_t)n * OUT_IMG;
    const int mb  = f0 + waveM * 32 + (lane >> 4) * 8;
    const int pcb = p0 + waveP * 32 + (lane & 15);
    if (pcb < P_IMG) {
#pragma unroll
        for (int r2 = 0; r2 < 8; ++r2) {
            outn[(size_t)(mb + r2)      * P_IMG + pcb] = acc00[r2];
            outn[(size_t)(mb + 16 + r2) * P_IMG + pcb] = acc10[r2];
        }
    }
    if (pcb + 16 < P_IMG) {
#pragma unroll
        for (int r2 = 0; r2 < 8; ++r2) {
            outn[(size_t)(mb + r2)      * P_IMG + pcb + 16] = acc01[r2];
            outn[(size_t)(mb + 16 + r2) * P_IMG + pcb + 16] = acc11[r2];
        }
    }
}

extern "C" void kernel_launch(void* const* d_in, const int* in_sizes, int n_in,
                              void* d_out, int out_size, void* d_ws, size_t ws_size,
                              hipStream_t stream) {
    const float* x = (const float*)d_in[0];   // (32,128,56,56) fp32
    const float* w = (const float*)d_in[1];   // (256,1152) fp32
    float* out = (float*)d_out;               // (32,256,54,54) fp32

    dim3 grid(F_OUT / M_BLK,                  // 2
              (P_IMG + P_BLK - 1) / P_BLK,    // 46
              32);
    dim3 block(256);
    sparse_conv2d_wmma_f32<<<grid, block, 0, stream>>>(x, w, out);
}